// OODDetector_80582176407863
// MI455X (gfx1250) — compile-verified
//
#include <hip/hip_runtime.h>
#include <math.h>

typedef float v2f __attribute__((ext_vector_type(2)));
typedef float v4f __attribute__((ext_vector_type(4)));
typedef float v8f __attribute__((ext_vector_type(8)));

#define DIMX 2048
#define SEQ  2048
#define BB   32
#define HH   256
#define KCEN 64

__device__ __forceinline__ float gelu_exact(float x) {
  return 0.5f * x * (1.0f + erff(x * 0.70710678118654752440f));
}
__device__ __forceinline__ float sigmoidf_(float x) {
  return 1.0f / (1.0f + expf(-x));
}

#if defined(__has_builtin)
#if __has_builtin(__builtin_amdgcn_wmma_f32_16x16x4_f32)
#define HAVE_WMMA_F32X4 1
#endif
#endif

__device__ __forceinline__ v8f wmma_f32_k4(v2f a, v2f b, v8f c) {
#ifdef HAVE_WMMA_F32X4
  return __builtin_amdgcn_wmma_f32_16x16x4_f32(false, a, false, b, (short)0, c,
                                               false, false);
#else
  // fallback only so the file still compiles if the builtin is absent
  c[0] += a.x * b.x + a.y * b.y;
  return c;
#endif
}

// ---------------------------------------------------------------------------
// K1a: partial pooling.  grid = B * 2(dtile) * 8(lseg) = 512 blocks, 256 thr.
// partial[(b*8+ls)*2048 + d] = sum over 256 L rows of x[b, l, d]
// ---------------------------------------------------------------------------
__global__ void pool_partial(const float* __restrict__ x,
                             float* __restrict__ partial) {
  int blk = blockIdx.x;
  int b  = blk >> 4;
  int dt = (blk >> 3) & 1;
  int ls = blk & 7;
  int d  = dt * 1024 + threadIdx.x * 4;
  const float* p = x + ((size_t)b * SEQ + (size_t)ls * 256) * DIMX + d;
  v4f acc = {0.f, 0.f, 0.f, 0.f};
  for (int l = 0; l < 256; ++l) {
    __builtin_prefetch(p + (size_t)(l + 16) * DIMX, 0, 1);
    acc += *reinterpret_cast<const v4f*>(p + (size_t)l * DIMX);
  }
  *reinterpret_cast<v4f*>(partial + ((size_t)(b * 8 + ls)) * DIMX + d) = acc;
}

// K1b: reduce partials -> pooled.  grid = 64 blocks, 256 thr.
__global__ void pool_reduce(const float* __restrict__ partial,
                            float* __restrict__ pooled) {
  int blk = blockIdx.x;
  int b  = blk >> 1;
  int dt = blk & 1;
  int d  = dt * 1024 + threadIdx.x * 4;
  v4f acc = {0.f, 0.f, 0.f, 0.f};
  for (int ls = 0; ls < 8; ++ls)
    acc += *reinterpret_cast<const v4f*>(partial + ((size_t)(b * 8 + ls)) * DIMX + d);
  acc *= (1.0f / (float)SEQ);
  *reinterpret_cast<v4f*>(pooled + (size_t)b * DIMX + d) = acc;
}

// ---------------------------------------------------------------------------
// K2: spectral-norm sigmas.  sigma = ||W @ l2norm(W^T u)||.  1 block, 1024 thr.
// sig[0] = sigma(W1), sig[1] = sigma(W2)
// ---------------------------------------------------------------------------
__global__ void sigma_kernel(const float* __restrict__ W1,
                             const float* __restrict__ u1,
                             const float* __restrict__ W2,
                             const float* __restrict__ u2,
                             float* __restrict__ sig) {
  __shared__ float vtmp[DIMX];
  __shared__ float red[1024];
  int t = threadIdx.x;

  // ---- W1: v = W1^T u1 ----
  float n2 = 0.f;
  for (int d = t; d < DIMX; d += 1024) {
    float s = 0.f;
    for (int h = 0; h < HH; ++h) s += W1[(size_t)h * DIMX + d] * u1[h];
    vtmp[d] = s;
    n2 += s * s;
  }
  red[t] = n2;
  __syncthreads();
  for (int o = 512; o > 0; o >>= 1) {
    if (t < o) red[t] += red[t + o];
    __syncthreads();
  }
  float inv = 1.0f / (sqrtf(red[0]) + 1e-12f);
  __syncthreads();
  // ---- sigma1^2 = sum_h ( (W1 v̂)[h] )^2 ----
  float acc = 0.f;
  for (int h = t; h < HH; h += 1024) {
    float s = 0.f;
    for (int d = 0; d < DIMX; ++d) s += W1[(size_t)h * DIMX + d] * vtmp[d];
    s *= inv;
    acc += s * s;
  }
  red[t] = acc;
  __syncthreads();
  for (int o = 512; o > 0; o >>= 1) {
    if (t < o) red[t] += red[t + o];
    __syncthreads();
  }
  if (t == 0) sig[0] = sqrtf(red[0]);
  __syncthreads();

  // ---- W2 (256x256) ----
  float m2 = 0.f;
  for (int d = t; d < HH; d += 1024) {
    float s = 0.f;
    for (int h = 0; h < HH; ++h) s += W2[h * HH + d] * u2[h];
    vtmp[d] = s;
    m2 += s * s;
  }
  red[t] = m2;
  __syncthreads();
  for (int o = 512; o > 0; o >>= 1) {
    if (t < o) red[t] += red[t + o];
    __syncthreads();
  }
  float inv2 = 1.0f / (sqrtf(red[0]) + 1e-12f);
  __syncthreads();
  float acc2 = 0.f;
  for (int h = t; h < HH; h += 1024) {
    float s = 0.f;
    for (int d = 0; d < HH; ++d) s += W2[h * HH + d] * vtmp[d];
    s *= inv2;
    acc2 += s * s;
  }
  red[t] = acc2;
  __syncthreads();
  for (int o = 512; o > 0; o >>= 1) {
    if (t < o) red[t] += red[t + o];
    __syncthreads();
  }
  if (t == 0) sig[1] = sqrtf(red[0]);
}

// ---------------------------------------------------------------------------
// K3: h1 = gelu(pooled @ W1^T / sigma1 + b1)  via V_WMMA_F32_16X16X4_F32.
// grid = 32 blocks (one wave / 16x16 tile), 32 threads.
// A layout: lane m=lane&15, K-pair = 2*(lane>>4).  B[k][n] = W1[n][k].
// C/D: vgpr j -> M = j + 8*(lane>>4), N = lane&15.
// ---------------------------------------------------------------------------
__global__ void gemm1_kernel(const float* __restrict__ pooled,
                             const float* __restrict__ W1,
                             const float* __restrict__ b1,
                             const float* __restrict__ sig,
                             float* __restrict__ h1) {
  int tile = blockIdx.x;         // 0..31
  int mt = tile >> 4;            // 0..1
  int nt = tile & 15;            // 0..15
  int lane = threadIdx.x;        // 0..31
  int lo = lane & 15;
  int hf = lane >> 4;
  const float* arow = pooled + (size_t)(mt * 16 + lo) * DIMX;
  const float* brow = W1 + (size_t)(nt * 16 + lo) * DIMX;
  v8f c = {0.f, 0.f, 0.f, 0.f, 0.f, 0.f, 0.f, 0.f};
  for (int k0 = 0; k0 < DIMX; k0 += 4) {
    v2f a = *reinterpret_cast<const v2f*>(arow + k0 + 2 * hf);
    v2f b = *reinterpret_cast<const v2f*>(brow + k0 + 2 * hf);
    c = wmma_f32_k4(a, b, c);
  }
  float invs = 1.0f / sig[0];
  int n = nt * 16 + lo;
  float bias = b1[n];
  for (int j = 0; j < 8; ++j) {
    int M = mt * 16 + j + 8 * hf;
    h1[M * HH + n] = gelu_exact(c[j] * invs + bias);
  }
}

// ---------------------------------------------------------------------------
// K4: everything after h1.  1 block, 256 thr (8 waves).
// GEMM2 (WMMA, 4 tiles/wave) -> RMSNorm -> centroid EMA -> mahalanobis ->
// energy head -> batch maxima -> scalar outputs.
// ---------------------------------------------------------------------------
__global__ void tail_kernel(const float* __restrict__ h1,
                            const float* __restrict__ W2,
                            const float* __restrict__ b2,
                            const float* __restrict__ sig,
                            const float* __restrict__ rms_w,
                            const float* __restrict__ centroids,
                            const float* __restrict__ prec,
                            const float* __restrict__ eW1,
                            const float* __restrict__ eb1,
                            const float* __restrict__ eW2,
                            const float* __restrict__ eb2,
                            float* __restrict__ newc,     // ws [64*256]
                            float* __restrict__ out_tail) {
  __shared__ float feat[BB][HH];     // 32 KB: h2, then normalized feat
  __shared__ float e1s[BB][HH / 2];  // 16 KB
  __shared__ int   nearest[BB];
  __shared__ int   cntk[KCEN];
  __shared__ float rowstat[BB];
  __shared__ float mahv[BB];
  __shared__ float normv[BB];
  __shared__ float escv[BB];
  __shared__ float mx[2];

  int tid  = threadIdx.x;
  int wave = tid >> 5;
  int lane = tid & 31;
  int lo = lane & 15;
  int hf = lane >> 4;
  float inv2 = 1.0f / sig[1];

  // ---- GEMM2: h2 = h1 @ W2^T / sigma2 + b2 ----
  for (int i = 0; i < 4; ++i) {
    int t  = wave * 4 + i;   // 0..31
    int mt = t >> 4;
    int nt = t & 15;
    const float* arow = h1 + (size_t)(mt * 16 + lo) * HH;
    const float* brow = W2 + (size_t)(nt * 16 + lo) * HH;
    v8f c = {0.f, 0.f, 0.f, 0.f, 0.f, 0.f, 0.f, 0.f};
    for (int k0 = 0; k0 < HH; k0 += 4) {
      v2f a = *reinterpret_cast<const v2f*>(arow + k0 + 2 * hf);
      v2f b = *reinterpret_cast<const v2f*>(brow + k0 + 2 * hf);
      c = wmma_f32_k4(a, b, c);
    }
    int n = nt * 16 + lo;
    float bias = b2[n];
    for (int j = 0; j < 8; ++j) {
      int M = mt * 16 + j + 8 * hf;
      feat[M][n] = c[j] * inv2 + bias;
    }
  }
  __syncthreads();

  // ---- RMSNorm ----
  if (tid < BB) {
    float ss = 0.f;
    for (int h = 0; h < HH; ++h) ss += feat[tid][h] * feat[tid][h];
    rowstat[tid] = rsqrtf(ss * (1.0f / (float)HH) + 1e-6f);
  }
  __syncthreads();
  for (int idx = tid; idx < BB * HH; idx += 256) {
    int r = idx >> 8, h = idx & 255;
    feat[r][h] = feat[r][h] * rowstat[r] * rms_w[h];
  }
  __syncthreads();

  // ---- nearest centroid (plain L2 to ORIGINAL centroids) ----
  if (tid < BB) {
    float best = 3.4e38f;
    int bi = 0;
    for (int k = 0; k < KCEN; ++k) {
      const float* cp = centroids + k * HH;
      float s = 0.f;
      for (int h = 0; h < HH; ++h) {
        float d = feat[tid][h] - cp[h];
        s += d * d;
      }
      if (s < best) { best = s; bi = k; }
    }
    nearest[tid] = bi;
  }
  __syncthreads();
  if (tid < KCEN) {
    int c = 0;
    for (int b = 0; b < BB; ++b) c += (nearest[b] == tid) ? 1 : 0;
    cntk[tid] = c;
  }
  __syncthreads();

  // ---- EMA update -> newc (global ws) ----
  for (int idx = tid; idx < KCEN * HH; idx += 256) {
    int k = idx >> 8, h = idx & 255;
    float cv = centroids[idx];
    int c = cntk[k];
    float nc = cv;
    if (c > 0) {
      float s = 0.f;
      for (int b = 0; b < BB; ++b)
        if (nearest[b] == k) s += feat[b][h];
      nc = 0.99f * cv + 0.01f * (s / (float)c);
    }
    newc[idx] = nc;
  }
  __syncthreads();

  // ---- mahalanobis min distance to updated centroids ----
  if (tid < BB) {
    float mn = 3.4e38f;
    for (int k = 0; k < KCEN; ++k) {
      const float* cp = newc + k * HH;
      float s = 0.f;
      for (int h = 0; h < HH; ++h) {
        float d = feat[tid][h] - cp[h];
        s += d * d * prec[h];
      }
      mn = fminf(mn, s);
    }
    mahv[tid] = sqrtf(mn);
  }

  // ---- energy head layer 1 ----
  for (int idx = tid; idx < BB * (HH / 2); idx += 256) {
    int b = idx >> 7, j = idx & 127;
    const float* w = eW1 + j * HH;
    float s = eb1[j];
    for (int h = 0; h < HH; ++h) s += feat[b][h] * w[h];
    e1s[b][j] = gelu_exact(s);
  }
  __syncthreads();
  if (tid < BB) {
    float s = eb2[0];
    for (int j = 0; j < HH / 2; ++j) s += e1s[tid][j] * eW2[j];
    escv[tid] = sigmoidf_(s);
    float nn = 0.f;
    for (int h = 0; h < HH; ++h) nn += feat[tid][h] * feat[tid][h];
    normv[tid] = sqrtf(nn);
  }
  __syncthreads();
  if (tid == 0) {
    float mm = 0.f, nm = 0.f;
    for (int b = 0; b < BB; ++b) {
      mm = fmaxf(mm, mahv[b]);
      nm = fmaxf(nm, normv[b]);
    }
    mx[0] = mm;
    mx[1] = nm;
  }
  __syncthreads();
  if (tid < BB) {
    float spec = 1.0f - normv[tid] / (mx[1] + 1e-6f);
    float ood = (mahv[tid] / (mx[0] + 1e-6f) + escv[tid] + spec) * (1.0f / 3.0f);
    out_tail[tid]          = ood;          // ood_score
    out_tail[32 + tid]     = mahv[tid];    // mah
    out_tail[64 + tid]     = escv[tid];    // energy_score
    out_tail[96 + tid]     = spec;         // spectral_unc
  }
}

// ---------------------------------------------------------------------------
// K5: gate + rescale stream.  grid = 32 * 2 * 16 = 1024 blocks, 256 thr.
// ---------------------------------------------------------------------------
__global__ void gate_kernel(const float* __restrict__ x,
                            const float* __restrict__ gW,
                            const float* __restrict__ gb,
                            const float* __restrict__ ood,  // out_tail[0..31]
                            float* __restrict__ xout) {
  int blk = blockIdx.x;
  int b  = blk >> 5;
  int dt = (blk >> 4) & 1;
  int ls = blk & 15;
  int d  = dt * 1024 + threadIdx.x * 4;
  float s = ood[b];
  v4f gw = *reinterpret_cast<const v4f*>(gW + d);
  v4f gbv = *reinterpret_cast<const v4f*>(gb + d);
  v4f scale;
  scale.x = 0.7f + 0.3f * sigmoidf_(s * gw.x + gbv.x);
  scale.y = 0.7f + 0.3f * sigmoidf_(s * gw.y + gbv.y);
  scale.z = 0.7f + 0.3f * sigmoidf_(s * gw.z + gbv.z);
  scale.w = 0.7f + 0.3f * sigmoidf_(s * gw.w + gbv.w);
  size_t base = ((size_t)b * SEQ + (size_t)ls * 128) * DIMX + d;
  const float* px = x + base;
  float* po = xout + base;
  for (int l = 0; l < 128; ++l) {
    __builtin_prefetch(px + (size_t)(l + 16) * DIMX, 0, 1);
    v4f v = *reinterpret_cast<const v4f*>(px + (size_t)l * DIMX);
    *reinterpret_cast<v4f*>(po + (size_t)l * DIMX) = v * scale;
  }
}

// ---------------------------------------------------------------------------
extern "C" void kernel_launch(void* const* d_in, const int* in_sizes, int n_in,
                              void* d_out, int out_size, void* d_ws,
                              size_t ws_size, hipStream_t stream) {
  const float* x         = (const float*)d_in[0];
  const float* W1        = (const float*)d_in[1];
  const float* b1        = (const float*)d_in[2];
  const float* u1        = (const float*)d_in[3];
  const float* W2        = (const float*)d_in[4];
  const float* b2        = (const float*)d_in[5];
  const float* u2        = (const float*)d_in[6];
  const float* rms_w     = (const float*)d_in[7];
  const float* centroids = (const float*)d_in[8];
  const float* prec      = (const float*)d_in[9];
  const float* eW1       = (const float*)d_in[10];
  const float* eb1       = (const float*)d_in[11];
  const float* eW2       = (const float*)d_in[12];
  const float* eb2       = (const float*)d_in[13];
  const float* gW        = (const float*)d_in[14];
  const float* gb        = (const float*)d_in[15];

  float* ws = (float*)d_ws;
  float* pooled  = ws;                       // 32*2048        = 65536
  float* h1      = ws + 65536;               // 32*256         = 8192
  float* sig     = ws + 73728;               // 2 (+pad to 16)
  float* newc    = ws + 73744;               // 64*256         = 16384
  float* partial = ws + 90128;               // 32*8*2048      = 524288

  float* out = (float*)d_out;
  float* out_tail = out + (size_t)BB * SEQ * DIMX;  // scores after x_ood

  pool_partial<<<512, 256, 0, stream>>>(x, partial);
  pool_reduce<<<64, 256, 0, stream>>>(partial, pooled);
  sigma_kernel<<<1, 1024, 0, stream>>>(W1, u1, W2, u2, sig);
  gemm1_kernel<<<32, 32, 0, stream>>>(pooled, W1, b1, sig, h1);
  tail_kernel<<<1, 256, 0, stream>>>(h1, W2, b2, sig, rms_w, centroids, prec,
                                     eW1, eb1, eW2, eb2, newc, out_tail);
  gate_kernel<<<1024, 256, 0, stream>>>(x, gW, gb, out_tail, out);
}